// GraphGNN_84954453115524
// MI455X (gfx1250) — compile-verified
//
#include <hip/hip_runtime.h>
#include <hip/hip_bf16.h>

typedef __attribute__((ext_vector_type(2))) float v2f;
typedef __attribute__((ext_vector_type(8))) float v8f;

#define N_GRAPHS 512
#define D_OUT    256
#define KPAD     64
#define K_REAL   58
#define EMB_D    50
#define L_WORDS  16

// ---------------------------------------------------------------------------
// 0. init: deg = 1.0 (self loop), pooling accumulators = 0
// ---------------------------------------------------------------------------
__global__ void init_buffers_kernel(float* __restrict__ deg, int N,
                                    float* __restrict__ gmax,
                                    float* __restrict__ gsum,
                                    float* __restrict__ cnt) {
    int i = blockIdx.x * blockDim.x + threadIdx.x;
    if (i < N) deg[i] = 1.0f;
    if (i < N_GRAPHS * D_OUT) { gmax[i] = 0.0f; gsum[i] = 0.0f; }
    if (i < N_GRAPHS) cnt[i] = 0.0f;
}

// ---------------------------------------------------------------------------
// 0b. pad gcn_w (58x256) -> (64x256), zero tail rows
// ---------------------------------------------------------------------------
__global__ void pad_w_kernel(const float* __restrict__ gcn_w,
                             float* __restrict__ wpad) {
    int i = blockIdx.x * blockDim.x + threadIdx.x;   // 0 .. 64*256-1
    if (i >= KPAD * D_OUT) return;
    int k = i / D_OUT, n = i % D_OUT;
    wpad[i] = (k < K_REAL) ? gcn_w[k * D_OUT + n] : 0.0f;
}

// ---------------------------------------------------------------------------
// 1. node features: embedding means + conv matvec + one-hot -> featPad (N x 64)
//    one block (64 threads) per node
// ---------------------------------------------------------------------------
__global__ void feature_kernel(const int* __restrict__ x,          // (N,17)
                               const float* __restrict__ word2vec, // (W,50)
                               const float* __restrict__ conv_w,   // (50,50,2)
                               const float* __restrict__ conv_b,   // (50)
                               float* __restrict__ featPad,        // (N,64)
                               int N) {
    int node = blockIdx.x;
    if (node >= N) return;
    int t = threadIdx.x;
    __shared__ float lr[2 * EMB_D];   // left[50] | right[50]

    if (t < EMB_D) {
        float s = 0.0f, e0 = 0.0f, e15 = 0.0f;
        #pragma unroll
        for (int w = 0; w < L_WORDS; ++w) {
            int idx = x[node * 17 + w];                       // 0..W ; 0 => zero row
            float v = (idx > 0) ? word2vec[(idx - 1) * EMB_D + t] : 0.0f;
            s += v;
            if (w == 0)  e0  = v;
            if (w == 15) e15 = v;
        }
        lr[t]         = (s - e15) * (1.0f / 15.0f);  // left mean (words 0..14)
        lr[EMB_D + t] = (s - e0)  * (1.0f / 15.0f);  // right mean (words 1..15)
    }
    __syncthreads();

    int ty = x[node * 17 + 16];
    bool valid = (ty >= 0);
    float* frow = featPad + (size_t)node * KPAD;

    if (t < EMB_D) {
        float acc = conv_b[t];
        #pragma unroll 5
        for (int k = 0; k < EMB_D; ++k) {
            acc += lr[k]         * conv_w[(t * EMB_D + k) * 2 + 0];
            acc += lr[EMB_D + k] * conv_w[(t * EMB_D + k) * 2 + 1];
        }
        frow[t] = valid ? acc : 0.0f;
    } else {
        int d = t - EMB_D;                 // 0..13
        float v = 0.0f;
        if (valid && d < 8) {
            int ct = ty > 7 ? 7 : ty;      // clip(type,0,7), ty>=0 here
            v = (ct == d) ? 1.0f : 0.0f;
        }
        frow[t] = v;                       // dims 50..57 one-hot, 58..63 zero pad
    }
}

// ---------------------------------------------------------------------------
// 2. h = featPad (N x 64) @ wpad (64 x 256) via V_WMMA_F32_16X16X4_F32
//    8 waves / block, one 16x16 output tile per wave
// ---------------------------------------------------------------------------
__global__ void gemm_wmma_kernel(const float* __restrict__ A,   // (N,64)
                                 const float* __restrict__ B,   // (64,256)
                                 float* __restrict__ H,         // (N,256)
                                 int mTiles) {                  // N/16
    int wave = threadIdx.x >> 5;
    int lane = threadIdx.x & 31;
    int tileId = blockIdx.x * 8 + wave;
    int mTile = tileId >> 4;            // 16 N-tiles per M-tile (256/16)
    int nTile = tileId & 15;
    if (mTile >= mTiles) return;

    int rowBase = mTile * 16;
    int colBase = nTile * 16;
    int lHalf = lane >> 4;              // 0: lanes 0-15, 1: lanes 16-31
    int lMod  = lane & 15;

    const float* Arow = A + (size_t)(rowBase + lMod) * KPAD;
    const float* Bcol = B + colBase + lMod;

    v8f c = {};
    #pragma unroll
    for (int k0 = 0; k0 < KPAD; k0 += 4) {
        v2f a, b;
        // A 16x4: lanes 0-15 hold K={k0,k0+1}, lanes 16-31 hold K={k0+2,k0+3}
        a.x = Arow[k0 + 2 * lHalf + 0];
        a.y = Arow[k0 + 2 * lHalf + 1];
        // B 4x16: row striped across lanes, K split across lane halves
        b.x = Bcol[(size_t)(k0 + 2 * lHalf + 0) * D_OUT];
        b.y = Bcol[(size_t)(k0 + 2 * lHalf + 1) * D_OUT];
        c = __builtin_amdgcn_wmma_f32_16x16x4_f32(false, a, false, b,
                                                  (short)0, c, false, false);
    }

    // C/D layout: VGPR r -> M = r + 8*lHalf, lane%16 -> column
    float* out = H + colBase + lMod;
    #pragma unroll
    for (int r = 0; r < 8; ++r) {
        int m = r + 8 * lHalf;
        out[(size_t)(rowBase + m) * D_OUT] = c[r];
    }
}

// ---------------------------------------------------------------------------
// 3. degree accumulation over edges (self loop already in init)
// ---------------------------------------------------------------------------
__global__ void degree_kernel(const int* __restrict__ col, float* __restrict__ deg,
                              int E) {
    int e = blockIdx.x * blockDim.x + threadIdx.x;
    if (e < E) atomicAdd(&deg[col[e]], 1.0f);
}

__global__ void rsqrt_kernel(float* __restrict__ deg, int N) {
    int i = blockIdx.x * blockDim.x + threadIdx.x;
    if (i < N) deg[i] = rsqrtf(deg[i]);     // deg >= 1 always
}

// ---------------------------------------------------------------------------
// 4. out1 init: self-loop term + bias.  One block (256 thr) per node.
// ---------------------------------------------------------------------------
__global__ void out1_init_kernel(const float* __restrict__ H,
                                 const float* __restrict__ dinv,
                                 const float* __restrict__ gcn_b,
                                 float* __restrict__ out1, int N) {
    int node = blockIdx.x;
    if (node >= N) return;
    int d = threadIdx.x;
    float di = dinv[node];
    size_t o = (size_t)node * D_OUT + d;
    out1[o] = H[o] * di * di + gcn_b[d];
}

// ---------------------------------------------------------------------------
// 5. edge scatter: out1[col] += h[row] * dinv[row]*dinv[col]
//    one wave per edge, 8 edges per block
// ---------------------------------------------------------------------------
__global__ void scatter_kernel(const int* __restrict__ row,
                               const int* __restrict__ col,
                               const float* __restrict__ H,
                               const float* __restrict__ dinv,
                               float* __restrict__ out1, int E) {
    int wave = threadIdx.x >> 5;
    int lane = threadIdx.x & 31;
    int e = blockIdx.x * 8 + wave;
    if (e >= E) return;
    int r = row[e], cIdx = col[e];
    float nrm = dinv[r] * dinv[cIdx];
    const float* hrow = H + (size_t)r * D_OUT;
    float* orow = out1 + (size_t)cIdx * D_OUT;
    #pragma unroll
    for (int j = 0; j < D_OUT / 32; ++j) {
        int d = lane + 32 * j;
        atomicAdd(&orow[d], hrow[d] * nrm);
    }
}

// ---------------------------------------------------------------------------
// 6. L2 normalize + ReLU + pooled segment max/sum/count. Wave per node.
// ---------------------------------------------------------------------------
__global__ void norm_pool_kernel(float* __restrict__ out1,
                                 const int* __restrict__ batch,
                                 float* __restrict__ gmax,
                                 float* __restrict__ gsum,
                                 float* __restrict__ cnt, int N) {
    int wave = threadIdx.x >> 5;
    int lane = threadIdx.x & 31;
    int node = blockIdx.x * 8 + wave;
    if (node >= N) return;

    float v[D_OUT / 32];
    float ss = 0.0f;
    size_t base = (size_t)node * D_OUT;
    #pragma unroll
    for (int j = 0; j < D_OUT / 32; ++j) {
        v[j] = out1[base + lane + 32 * j];
        ss += v[j] * v[j];
    }
    #pragma unroll
    for (int off = 16; off > 0; off >>= 1)
        ss += __shfl_xor(ss, off, 32);

    float inv = 1.0f / fmaxf(sqrtf(ss), 1e-12f);
    int b = batch[node];
    float* gm = gmax + (size_t)b * D_OUT;
    float* gs = gsum + (size_t)b * D_OUT;
    #pragma unroll
    for (int j = 0; j < D_OUT / 32; ++j) {
        float rlu = fmaxf(v[j] * inv, 0.0f);   // >= 0 -> int max == float max
        int d = lane + 32 * j;
        atomicAdd(&gs[d], rlu);
        atomicMax((int*)&gm[d], __float_as_int(rlu));
    }
    if (lane == 0) atomicAdd(&cnt[b], 1.0f);
}

// ---------------------------------------------------------------------------
// 7. final: [gmax | gsum/cnt] (512 x 512) @ lin_w.T (2 x 512) + lin_b
//    wave per graph
// ---------------------------------------------------------------------------
__global__ void final_kernel(const float* __restrict__ gmax,
                             const float* __restrict__ gsum,
                             const float* __restrict__ cnt,
                             const float* __restrict__ lin_w,  // (2,512)
                             const float* __restrict__ lin_b,  // (2)
                             float* __restrict__ out) {        // (512,2)
    int wave = threadIdx.x >> 5;
    int lane = threadIdx.x & 31;
    int g = blockIdx.x * 8 + wave;
    if (g >= N_GRAPHS) return;

    float ic = 1.0f / fmaxf(cnt[g], 1.0f);
    float a0 = 0.0f, a1 = 0.0f;
    #pragma unroll
    for (int j = 0; j < D_OUT / 32; ++j) {
        int d = lane + 32 * j;
        float pm = gmax[(size_t)g * D_OUT + d];
        float pe = gsum[(size_t)g * D_OUT + d] * ic;
        a0 += pm * lin_w[d]       + pe * lin_w[D_OUT + d];
        a1 += pm * lin_w[512 + d] + pe * lin_w[512 + D_OUT + d];
    }
    #pragma unroll
    for (int off = 16; off > 0; off >>= 1) {
        a0 += __shfl_xor(a0, off, 32);
        a1 += __shfl_xor(a1, off, 32);
    }
    if (lane == 0) {
        out[g * 2 + 0] = a0 + lin_b[0];
        out[g * 2 + 1] = a1 + lin_b[1];
    }
}

// ---------------------------------------------------------------------------
static inline size_t alignUp(size_t x) { return (x + 255) & ~(size_t)255; }

extern "C" void kernel_launch(void* const* d_in, const int* in_sizes, int n_in,
                              void* d_out, int out_size, void* d_ws, size_t ws_size,
                              hipStream_t stream) {
    const int*   x        = (const int*)  d_in[0];
    const int*   edge     = (const int*)  d_in[1];
    const int*   batch    = (const int*)  d_in[2];
    const float* word2vec = (const float*)d_in[3];
    const float* conv_w   = (const float*)d_in[4];
    const float* conv_b   = (const float*)d_in[5];
    const float* gcn_w    = (const float*)d_in[6];
    const float* gcn_b    = (const float*)d_in[7];
    const float* lin_w    = (const float*)d_in[8];
    const float* lin_b    = (const float*)d_in[9];
    float* out = (float*)d_out;

    const int N = in_sizes[2];          // 100000
    const int E = in_sizes[1] / 2;      // 800000
    const int* erow = edge;
    const int* ecol = edge + E;

    // workspace carve
    char* ws = (char*)d_ws;
    size_t off = 0;
    float* featPad = (float*)(ws + off); off = alignUp(off + (size_t)N * KPAD * 4);
    float* wpad    = (float*)(ws + off); off = alignUp(off + (size_t)KPAD * D_OUT * 4);
    float* H       = (float*)(ws + off); off = alignUp(off + (size_t)N * D_OUT * 4);
    float* dinv    = (float*)(ws + off); off = alignUp(off + (size_t)N * 4);
    float* out1    = (float*)(ws + off); off = alignUp(off + (size_t)N * D_OUT * 4);
    float* gmax    = (float*)(ws + off); off = alignUp(off + (size_t)N_GRAPHS * D_OUT * 4);
    float* gsum    = (float*)(ws + off); off = alignUp(off + (size_t)N_GRAPHS * D_OUT * 4);
    float* cnt     = (float*)(ws + off); off = alignUp(off + (size_t)N_GRAPHS * 4);
    (void)ws_size;

    // 0. init accumulators (deg lives in dinv buffer first)
    {
        int total = N;  // N > 512*256
        init_buffers_kernel<<<(total + 255) / 256, 256, 0, stream>>>(dinv, N, gmax, gsum, cnt);
        pad_w_kernel<<<(KPAD * D_OUT + 255) / 256, 256, 0, stream>>>(gcn_w, wpad);
    }
    // 1. node features
    feature_kernel<<<N, 64, 0, stream>>>(x, word2vec, conv_w, conv_b, featPad, N);
    // 2. GEMM (N is a multiple of 16: 100000 = 6250*16)
    {
        int mTiles = N / 16;
        int tiles = mTiles * (D_OUT / 16);
        gemm_wmma_kernel<<<(tiles + 7) / 8, 256, 0, stream>>>(featPad, wpad, H, mTiles);
    }
    // 3. degrees -> rsqrt
    degree_kernel<<<(E + 255) / 256, 256, 0, stream>>>(ecol, dinv, E);
    rsqrt_kernel<<<(N + 255) / 256, 256, 0, stream>>>(dinv, N);
    // 4. out1 init (self loop + bias)
    out1_init_kernel<<<N, D_OUT, 0, stream>>>(H, dinv, gcn_b, out1, N);
    // 5. edge scatter
    scatter_kernel<<<(E + 7) / 8, 256, 0, stream>>>(erow, ecol, H, dinv, out1, E);
    // 6. normalize + relu + pooling
    norm_pool_kernel<<<(N + 7) / 8, 256, 0, stream>>>(out1, batch, gmax, gsum, cnt, N);
    // 7. final linear
    final_kernel<<<(N_GRAPHS + 7) / 8, 256, 0, stream>>>(gmax, gsum, cnt, lin_w, lin_b, out);
}